// SMPL_37778532336267
// MI455X (gfx1250) — compile-verified
//
#include <hip/hip_runtime.h>
#include <hip/hip_bf16.h>
#include <math.h>

// ---------------- problem constants ----------------
#define NBATCH 1024
#define NVERTS 6890
#define NELEM  (NVERTS * 3)   // 20670
#define NELEMP 20672          // padded element dim (multiple of 32)
#define NJ     24
#define NPOSE  207
#define NPOSEP 208            // padded K for WMMA
#define NBETA  10
#define NBETAP 12             // padded K for WMMA
#define NCOCO  19
#define KBP    (NPOSEP / 4)   // 52 pose K-blocks
#define KBS    (NBETAP / 4)   // 3 shape K-blocks

typedef __attribute__((ext_vector_type(2))) float v2f;
typedef __attribute__((ext_vector_type(8))) float v8f;

#define WMMA_F32(a, b, c) \
    __builtin_amdgcn_wmma_f32_16x16x4_f32(false, (a), false, (b), (short)0, (c), false, false)

// ---------------------------------------------------------------------------
// Kernel 1: batch Rodrigues -> Rs (n,24,9) and padded pose_feature (n,208)
// ---------------------------------------------------------------------------
__global__ __launch_bounds__(256) void k_rodrigues(
    const float* __restrict__ theta,   // (1024, 72)
    float* __restrict__ Rs,            // ws (1024, 24, 9)
    float* __restrict__ pf)            // ws (1024, 208), zero padded
{
    int idx = blockIdx.x * blockDim.x + threadIdx.x;
    if (idx >= NBATCH * NJ) return;
    int n = idx / NJ, j = idx % NJ;

    float tx = theta[n * 72 + 3 * j + 0];
    float ty = theta[n * 72 + 3 * j + 1];
    float tz = theta[n * 72 + 3 * j + 2];
    float angle = sqrtf(tx * tx + ty * ty + tz * tz + 1e-8f);
    float inv = 1.0f / angle;
    float rx = tx * inv, ry = ty * inv, rz = tz * inv;
    float c = cosf(angle), s = sinf(angle), omc = 1.0f - c;

    float R[9];
    R[0] = c + omc * rx * rx;       R[1] = -s * rz + omc * rx * ry;  R[2] =  s * ry + omc * rx * rz;
    R[3] =  s * rz + omc * ry * rx; R[4] = c + omc * ry * ry;        R[5] = -s * rx + omc * ry * rz;
    R[6] = -s * ry + omc * rz * rx; R[7] =  s * rx + omc * rz * ry;  R[8] = c + omc * rz * rz;

    #pragma unroll
    for (int k = 0; k < 9; ++k) Rs[idx * 9 + k] = R[k];

    if (j >= 1) {
        #pragma unroll
        for (int k = 0; k < 9; ++k) {
            float e = ((k == 0) | (k == 4) | (k == 8)) ? 1.0f : 0.0f;
            pf[n * NPOSEP + (j - 1) * 9 + k] = R[k] - e;
        }
    } else {
        pf[n * NPOSEP + NPOSE] = 0.0f;   // zero the K-pad column
    }
}

// ---------------------------------------------------------------------------
// Pack kernels: build zero-padded, fragment-friendly B operands + padded beta.
// B layout: [kblock][e (20672)][4 k-values] -> each lane does ONE b64 load.
// ---------------------------------------------------------------------------
__global__ __launch_bounds__(256) void k_pack_B(
    const float* __restrict__ src, float* __restrict__ dst,
    int kmax, int nkb)     // src (kmax, 20670); dst (nkb, 20672, 4)
{
    int idx = blockIdx.x * blockDim.x + threadIdx.x;
    int total = nkb * NELEMP * 4;
    if (idx >= total) return;
    int kb  = idx / (NELEMP * 4);
    int rem = idx % (NELEMP * 4);
    int e   = rem >> 2;
    int kk  = rem & 3;
    int k   = kb * 4 + kk;
    dst[idx] = (k < kmax && e < NELEM) ? src[(size_t)k * NELEM + e] : 0.0f;
}

__global__ __launch_bounds__(256) void k_pack_beta(
    const float* __restrict__ beta, float* __restrict__ betap)  // (1024,10)->(1024,12)
{
    int idx = blockIdx.x * blockDim.x + threadIdx.x;
    if (idx >= NBATCH * NBETAP) return;
    int n = idx / NBETAP, k = idx % NBETAP;
    betap[idx] = (k < NBETA) ? beta[n * NBETA + k] : 0.0f;
}

// ---------------------------------------------------------------------------
// Kernel 2: fused shape+pose blend GEMMs via V_WMMA_F32_16X16X4_F32.
//   v_shaped = beta @ shapedirs + v_template          (K = 12 padded)
//   v_posed  = v_shaped + pose_feature @ posedirs     (K = 208 padded)
// One 16(batch) x 32(element) output per wave (two 16x16 tiles, A reused),
// 8 waves/block, fully unrolled K loop, two interleaved accumulators.
// All loads unconditional (operands pre-padded) -> EXEC stays all-ones.
// ---------------------------------------------------------------------------
__global__ __launch_bounds__(256) void k_blend_wmma(
    const float* __restrict__ betap,      // ws (1024, 12)
    const float* __restrict__ v_template, // (20670)
    const float* __restrict__ Bshape,     // ws (3, 20672, 4)
    const float* __restrict__ Bpose,      // ws (52, 20672, 4)
    const float* __restrict__ pf,         // ws (1024, 208)
    float* __restrict__ v_shaped,         // ws (1024, 20672)
    float* __restrict__ v_posed)          // ws (1024, 20672)
{
    const int lane = threadIdx.x & 31;
    const int wave = threadIdx.x >> 5;
    const int ntilesN2 = NELEMP / 32;              // 646
    const int tile = blockIdx.x * 8 + wave;        // grid sized exactly
    const int n0 = (tile / ntilesN2) * 16;
    const int e0 = (tile % ntilesN2) * 32;

    const int mrow = n0 + (lane & 15);             // A row (batch)
    const int kh   = lane >> 4;                    // K half: 0 or 1
    const int ec0  = e0 + (lane & 15);             // first tile column
    const int ec1  = ec0 + 16;                     // second tile column

    // ---- pose GEMM, K = 208, fully unrolled, 2 accумulators per tile ----
    const v2f* __restrict__ ap = (const v2f*)(pf + (size_t)mrow * NPOSEP);
    const v2f* __restrict__ bp = (const v2f*)Bpose;
    v8f c0a = {}, c0b = {}, c1a = {}, c1b = {};
    #pragma unroll
    for (int kb = 0; kb < KBP; ++kb) {
        const v2f a  = ap[kb * 2 + kh];
        const v2f b0 = bp[((size_t)kb * NELEMP + ec0) * 2 + kh];
        const v2f b1 = bp[((size_t)kb * NELEMP + ec1) * 2 + kh];
        if (kb & 1) {
            c0b = WMMA_F32(a, b0, c0b);
            c1b = WMMA_F32(a, b1, c1b);
        } else {
            c0a = WMMA_F32(a, b0, c0a);
            c1a = WMMA_F32(a, b1, c1a);
        }
    }
    const v8f cp0 = c0a + c0b;
    const v8f cp1 = c1a + c1b;

    // ---- shape GEMM, K = 12 ----
    const v2f* __restrict__ as = (const v2f*)(betap + (size_t)mrow * NBETAP);
    const v2f* __restrict__ bs = (const v2f*)Bshape;
    v8f s0 = {}, s1 = {};
    #pragma unroll
    for (int kb = 0; kb < KBS; ++kb) {
        const v2f a  = as[kb * 2 + kh];
        const v2f b0 = bs[((size_t)kb * NELEMP + ec0) * 2 + kh];
        const v2f b1 = bs[((size_t)kb * NELEMP + ec1) * 2 + kh];
        s0 = WMMA_F32(a, b0, s0);
        s1 = WMMA_F32(a, b1, s1);
    }

    // ---- epilogue: + v_template, write v_shaped & v_posed (padded rows) ----
    const float vt0 = (ec0 < NELEM) ? v_template[ec0] : 0.0f;
    const float vt1 = (ec1 < NELEM) ? v_template[ec1] : 0.0f;
    #pragma unroll
    for (int r = 0; r < 8; ++r) {
        const int row = n0 + r + 8 * kh;           // D: M = r + 8*(lane>>4)
        const size_t base = (size_t)row * NELEMP;
        const float vs0 = s0[r] + vt0;
        const float vs1 = s1[r] + vt1;
        v_shaped[base + ec0] = vs0;
        v_shaped[base + ec1] = vs1;
        v_posed [base + ec0] = vs0 + cp0[r];
        v_posed [base + ec1] = vs1 + cp1[r];
    }
}

// ---------------------------------------------------------------------------
// Kernel 3: J[n,j,a] = sum_v v_shaped[n,v,a] * J_regressor[v,j]
// One block per batch; 72 register accumulators/thread; LDS f32-atomic reduce.
// ---------------------------------------------------------------------------
__global__ __launch_bounds__(256) void k_jreg(
    const float* __restrict__ v_shaped,   // ws (1024, 20672)
    const float* __restrict__ Jreg,       // (6890, 24)
    float* __restrict__ J)                // ws (1024, 24, 3)
{
    __shared__ float sacc[NJ * 3];
    const int n = blockIdx.x;
    for (int i = threadIdx.x; i < NJ * 3; i += blockDim.x) sacc[i] = 0.0f;
    __syncthreads();

    float acc[NJ][3];
    #pragma unroll
    for (int j = 0; j < NJ; ++j) { acc[j][0] = acc[j][1] = acc[j][2] = 0.0f; }

    const float* vsn = v_shaped + (size_t)n * NELEMP;
    for (int v = threadIdx.x; v < NVERTS; v += blockDim.x) {
        const float px = vsn[3 * v + 0];
        const float py = vsn[3 * v + 1];
        const float pz = vsn[3 * v + 2];
        #pragma unroll
        for (int j = 0; j < NJ; ++j) {
            const float w = Jreg[v * NJ + j];
            acc[j][0] += w * px; acc[j][1] += w * py; acc[j][2] += w * pz;
        }
    }
    #pragma unroll
    for (int j = 0; j < NJ; ++j) {
        atomicAdd(&sacc[j * 3 + 0], acc[j][0]);
        atomicAdd(&sacc[j * 3 + 1], acc[j][1]);
        atomicAdd(&sacc[j * 3 + 2], acc[j][2]);
    }
    __syncthreads();
    if (threadIdx.x < NJ * 3) J[n * NJ * 3 + threadIdx.x] = sacc[threadIdx.x];
}

// ---------------------------------------------------------------------------
// Kernel 4: kinematic chain. One thread per batch -> A (1024,24,3x4)
// with init_bone folded: A.t = W.t - W.R @ J_i
// ---------------------------------------------------------------------------
__global__ __launch_bounds__(256) void k_chain(
    const float* __restrict__ Rs,   // ws (1024, 24, 9)
    const float* __restrict__ J,    // ws (1024, 24, 3)
    float* __restrict__ A)          // ws (1024, 24, 12)
{
    const int n = blockIdx.x * blockDim.x + threadIdx.x;
    if (n >= NBATCH) return;
    const int PAR[NJ] = {0,0,0,0,1,2,3,4,5,6,7,8,9,9,9,12,13,14,16,17,18,19,20,21};

    float W[NJ][12];
    for (int i = 0; i < NJ; ++i) {
        const float* R  = Rs + ((size_t)n * NJ + i) * 9;
        const float jx = J[(n * NJ + i) * 3 + 0];
        const float jy = J[(n * NJ + i) * 3 + 1];
        const float jz = J[(n * NJ + i) * 3 + 2];
        if (i == 0) {
            #pragma unroll
            for (int r = 0; r < 3; ++r) {
                W[0][r * 4 + 0] = R[r * 3 + 0];
                W[0][r * 4 + 1] = R[r * 3 + 1];
                W[0][r * 4 + 2] = R[r * 3 + 2];
                W[0][r * 4 + 3] = J[(n * NJ) * 3 + r];
            }
        } else {
            const int p = PAR[i];
            const float tx = jx - J[(n * NJ + p) * 3 + 0];
            const float ty = jy - J[(n * NJ + p) * 3 + 1];
            const float tz = jz - J[(n * NJ + p) * 3 + 2];
            #pragma unroll
            for (int r = 0; r < 3; ++r) {
                const float p0 = W[p][r * 4 + 0], p1 = W[p][r * 4 + 1];
                const float p2 = W[p][r * 4 + 2], p3 = W[p][r * 4 + 3];
                W[i][r * 4 + 0] = p0 * R[0] + p1 * R[3] + p2 * R[6];
                W[i][r * 4 + 1] = p0 * R[1] + p1 * R[4] + p2 * R[7];
                W[i][r * 4 + 2] = p0 * R[2] + p1 * R[5] + p2 * R[8];
                W[i][r * 4 + 3] = p0 * tx + p1 * ty + p2 * tz + p3;
            }
        }
        float* Ao = A + ((size_t)n * NJ + i) * 12;
        #pragma unroll
        for (int r = 0; r < 3; ++r) {
            const float w0 = W[i][r * 4 + 0], w1 = W[i][r * 4 + 1], w2 = W[i][r * 4 + 2];
            Ao[r * 4 + 0] = w0;
            Ao[r * 4 + 1] = w1;
            Ao[r * 4 + 2] = w2;
            Ao[r * 4 + 3] = W[i][r * 4 + 3] - (w0 * jx + w1 * jy + w2 * jz);
        }
    }
}

// ---------------------------------------------------------------------------
// Kernel 5: fused LBS skinning + COCO joint regression (verts never hit HBM).
// ---------------------------------------------------------------------------
__global__ __launch_bounds__(256) void k_skin_coco(
    const float* __restrict__ v_posed,  // ws (1024, 20672)
    const float* __restrict__ weights,  // (6890, 24)
    const float* __restrict__ A,        // ws (1024, 24, 12)
    const float* __restrict__ cocoR,    // (6890, 19)
    float* __restrict__ joints)         // d_out (1024, 19, 3)
{
    __shared__ float sA[NJ * 12];
    __shared__ float sacc[NCOCO * 3];
    const int n = blockIdx.x;
    for (int i = threadIdx.x; i < NJ * 12; i += blockDim.x) sA[i] = A[(size_t)n * NJ * 12 + i];
    for (int i = threadIdx.x; i < NCOCO * 3; i += blockDim.x) sacc[i] = 0.0f;
    __syncthreads();

    float acc[NCOCO][3];
    #pragma unroll
    for (int c = 0; c < NCOCO; ++c) { acc[c][0] = acc[c][1] = acc[c][2] = 0.0f; }

    const float* vpn = v_posed + (size_t)n * NELEMP;
    for (int v = threadIdx.x; v < NVERTS; v += blockDim.x) {
        const float px = vpn[3 * v + 0];
        const float py = vpn[3 * v + 1];
        const float pz = vpn[3 * v + 2];

        float T[12];
        #pragma unroll
        for (int k = 0; k < 12; ++k) T[k] = 0.0f;
        #pragma unroll
        for (int j = 0; j < NJ; ++j) {
            const float w = weights[v * NJ + j];
            #pragma unroll
            for (int k = 0; k < 12; ++k) T[k] += w * sA[j * 12 + k];
        }
        const float vx = T[0] * px + T[1] * py + T[2]  * pz + T[3];
        const float vy = T[4] * px + T[5] * py + T[6]  * pz + T[7];
        const float vz = T[8] * px + T[9] * py + T[10] * pz + T[11];

        #pragma unroll
        for (int c = 0; c < NCOCO; ++c) {
            const float w = cocoR[v * NCOCO + c];
            acc[c][0] += w * vx; acc[c][1] += w * vy; acc[c][2] += w * vz;
        }
    }
    #pragma unroll
    for (int c = 0; c < NCOCO; ++c) {
        atomicAdd(&sacc[c * 3 + 0], acc[c][0]);
        atomicAdd(&sacc[c * 3 + 1], acc[c][1]);
        atomicAdd(&sacc[c * 3 + 2], acc[c][2]);
    }
    __syncthreads();
    if (threadIdx.x < NCOCO * 3) joints[n * NCOCO * 3 + threadIdx.x] = sacc[threadIdx.x];
}

// ---------------------------------------------------------------------------
extern "C" void kernel_launch(void* const* d_in, const int* in_sizes, int n_in,
                              void* d_out, int out_size, void* d_ws, size_t ws_size,
                              hipStream_t stream) {
    const float* beta       = (const float*)d_in[0];  // (1024,10)
    const float* theta      = (const float*)d_in[1];  // (1024,72)
    const float* v_template = (const float*)d_in[2];  // (6890,3)
    const float* shapedirs  = (const float*)d_in[3];  // (10,20670)
    const float* Jreg       = (const float*)d_in[4];  // (6890,24)
    const float* posedirs   = (const float*)d_in[5];  // (207,20670)
    const float* weights    = (const float*)d_in[6];  // (6890,24)
    const float* cocoR      = (const float*)d_in[7];  // (6890,19)
    float* joints = (float*)d_out;                    // (1024,19,3)

    // workspace layout (floats); total ~47.7M floats = ~191 MB
    float* ws = (float*)d_ws;
    float* pf       = ws;                                    // 1024*208
    float* Rs       = pf     + (size_t)NBATCH * NPOSEP;
    float* J        = Rs     + (size_t)NBATCH * NJ * 9;
    float* Amat     = J      + (size_t)NBATCH * NJ * 3;
    float* betap    = Amat   + (size_t)NBATCH * NJ * 12;
    float* Bshape   = betap  + (size_t)NBATCH * NBETAP;      // 3*20672*4
    float* Bpose    = Bshape + (size_t)KBS * NELEMP * 4;     // 52*20672*4
    float* v_shaped = Bpose  + (size_t)KBP * NELEMP * 4;     // 1024*20672
    float* v_posed  = v_shaped + (size_t)NBATCH * NELEMP;

    // 1. Rodrigues + pose_feature (padded)
    k_rodrigues<<<(NBATCH * NJ + 255) / 256, 256, 0, stream>>>(theta, Rs, pf);

    // 1b. pack GEMM operands (zero-padded, b64-fragment layout)
    k_pack_B<<<(KBP * NELEMP * 4 + 255) / 256, 256, 0, stream>>>(posedirs,  Bpose,  NPOSE, KBP);
    k_pack_B<<<(KBS * NELEMP * 4 + 255) / 256, 256, 0, stream>>>(shapedirs, Bshape, NBETA, KBS);
    k_pack_beta<<<(NBATCH * NBETAP + 255) / 256, 256, 0, stream>>>(beta, betap);

    // 2. WMMA blend GEMMs: 64 x 646 (16x32) tiles, 8 waves/block, exact grid
    const int ntiles = (NBATCH / 16) * (NELEMP / 32);        // 41344
    k_blend_wmma<<<ntiles / 8, 256, 0, stream>>>(
        betap, v_template, Bshape, Bpose, pf, v_shaped, v_posed);

    // 3. joint locations from shaped mesh
    k_jreg<<<NBATCH, 256, 0, stream>>>(v_shaped, Jreg, J);

    // 4. kinematic chain -> skinning transforms
    k_chain<<<(NBATCH + 255) / 256, 256, 0, stream>>>(Rs, J, Amat);

    // 5. fused skinning + COCO joint regression
    k_skin_coco<<<NBATCH, 256, 0, stream>>>(v_posed, weights, Amat, cocoR, joints);
}